// FieldOfJunctions_52329881534949
// MI455X (gfx1250) — compile-verified
//
#include <hip/hip_runtime.h>
#include <math.h>

#define PSZ 8
#define HPP 57
#define WPP 57
#define NV  31
#define BB  2
#define CINN 32
#define HH  64
#define WW  64
#define NPATCH (BB*HPP*WPP)   /* 6498 */

#define TWO_PI_F 6.2831853071795864769f
#define PI_F     3.1415926535897932385f
#define ETA_F    0.01f
#define TAU_F    0.1f
#define DELTA_F  0.05f

// padded row stride for the unified 16-row LDS matrix: 68 floats = 272 B
// => per-row bank rotation of 4, so 16 lanes reading 16 rows at the same
// column offset touch 16 distinct banks (stride 64 would be a 16-way conflict)
#define ABSTR 68

typedef __attribute__((ext_vector_type(2))) float v2f;
typedef __attribute__((ext_vector_type(8))) float v8f;

// ---------------------------------------------------------------- helpers
__device__ __forceinline__ float pymod2pi(float x) {
    float r = fmodf(x, TWO_PI_F);
    if (r < 0.f) r += TWO_PI_F;
    return r;
}
__device__ __forceinline__ float pow35(float t) {
    float t2 = t * t, t4 = t2 * t2, t8 = t4 * t4, t16 = t8 * t8, t32 = t16 * t16;
    return t32 * t2 * t;
}
__device__ __forceinline__ float siluf(float x) {
    return x / (1.f + expf(-x));
}
__device__ __forceinline__ float wred(float v) {
#pragma unroll
    for (int o = 16; o > 0; o >>= 1) v += __shfl_xor(v, o, 32);
    return v;  // xor-reduce: every lane holds the total
}

struct Geom {
    float sa1, ca1, sa2, ca2, sa3, ca3, sa4, ca4;
    float x0, y0, s13, s42, gt13, gt42;
};

__device__ __forceinline__ void make_geom(float A0, float A1, float A2,
                                          float x0, float y0, Geom& G) {
    float a1 = pymod2pi(A0), a2 = pymod2pi(A1), a3 = pymod2pi(A2), t;
    if (a1 > a2) { t = a1; a1 = a2; a2 = t; }
    if (a2 > a3) { t = a2; a2 = a3; a3 = t; }
    if (a1 > a2) { t = a1; a1 = a2; a2 = t; }
    float hm = pymod2pi(0.5f * (a1 - a3));
    float a4 = 0.5f * (a1 + a3) + ((hm >= PI_F) ? PI_F : 0.f);
    float d42 = pymod2pi(a2 - a4);
    float d13 = pymod2pi(a3 - a1);
    G.s42 = (d42 < PI_F) ? 1.f : -1.f;
    G.s13 = (d13 < PI_F) ? 1.f : -1.f;
    G.sa1 = sinf(a1); G.ca1 = cosf(a1);
    G.sa2 = sinf(a2); G.ca2 = cosf(a2);
    G.sa3 = sinf(a3); G.ca3 = cosf(a3);
    G.sa4 = sinf(a4); G.ca4 = cosf(a4);
    G.x0 = x0; G.y0 = y0;
    G.gt13 = pow35(d13 / PI_F - 1.f) * TAU_F;
    G.gt42 = pow35(d42 / PI_F - 1.f) * TAU_F;
}

__device__ __forceinline__ void dists_at(const Geom& G, float xg, float yg,
                                         float& d13, float& d42) {
    float dx = xg - G.x0, dy = yg - G.y0;
    float L1 = -G.sa1 * dx + G.ca1 * dy;
    float L2 = -G.sa2 * dx + G.ca2 * dy;
    float L3 = -G.sa3 * dx + G.ca3 * dy;
    float L4 = -G.sa4 * dx + G.ca4 * dy;
    d13 = G.s13 * fminf(G.s13 * L1, -G.s13 * L3) + G.gt13;
    d42 = G.s42 * fminf(G.s42 * L4, -G.s42 * L2) + G.gt42;
}
__device__ __forceinline__ void wedges_at(float d13, float d42,
                                          float& w0, float& w1, float& w2) {
    float h0 = 0.5f * (1.f + (2.f / PI_F) * atanf(d13 / ETA_F));
    float h1 = 0.5f * (1.f + (2.f / PI_F) * atanf(d42 / ETA_F));
    w0 = 1.f - h0;
    w1 = h0 * (1.f - h1);
    w2 = h0 * h1;
}

// ---------------------------------------------------------------- zero params
__global__ void k_zero(float* __restrict__ p, int n) {
    int i = blockIdx.x * blockDim.x + threadIdx.x;
    if (i < n) p[i] = 0.f;
}

// ---------------------------------------------------------------- conv1x1 + BN partial stats
__global__ void k_conv(const float* __restrict__ x, const float* __restrict__ rw,
                       float* __restrict__ y, float* __restrict__ part1) {
    __shared__ float srw[3][CINN];
    int tid = threadIdx.x;
    if (tid < 96) srw[tid / 32][tid % 32] = rw[tid];
    __syncthreads();

    int idx = blockIdx.x * 256 + tid;         // 0..8191 over (b,h,w)
    int b = idx >> 12;
    int hw = idx & 4095;
    const float* xb = x + (size_t)b * CINN * 4096 + hw;
    float o0 = 0.f, o1 = 0.f, o2 = 0.f;
#pragma unroll 8
    for (int c = 0; c < CINN; ++c) {
        float v = xb[(size_t)c * 4096];
        o0 += v * srw[0][c];
        o1 += v * srw[1][c];
        o2 += v * srw[2][c];
    }
    y[(size_t)(b * 3 + 0) * 4096 + hw] = o0;
    y[(size_t)(b * 3 + 1) * 4096 + hw] = o1;
    y[(size_t)(b * 3 + 2) * 4096 + hw] = o2;

    __shared__ float red[256];
    float vals[6] = { o0, o1, o2, o0 * o0, o1 * o1, o2 * o2 };
#pragma unroll
    for (int s = 0; s < 6; ++s) {
        red[tid] = vals[s];
        __syncthreads();
        for (int off = 128; off > 0; off >>= 1) {
            if (tid < off) red[tid] += red[tid + off];
            __syncthreads();
        }
        if (tid == 0) part1[blockIdx.x * 8 + s] = red[0];
        __syncthreads();
    }
}

// ---------------------------------------------------------------- tiny deterministic reduce
__global__ void k_red(const float* __restrict__ part, float* __restrict__ stats,
                      int nblocks, int nvals, int stride) {
    if (blockIdx.x == 0 && threadIdx.x == 0) {
        for (int v = 0; v < nvals; ++v) {
            float s = 0.f;
            for (int b = 0; b < nblocks; ++b) s += part[b * stride + v];
            stats[v] = s;
        }
    }
}

// ---------------------------------------------------------------- BN + SiLU (in place on y -> xr)
__global__ void k_bnsilu(float* __restrict__ y, const float* __restrict__ stats6,
                         const float* __restrict__ gamma, const float* __restrict__ beta) {
    int idx = blockIdx.x * blockDim.x + threadIdx.x;  // 0..24575
    if (idx >= BB * 3 * HH * WW) return;
    int c = (idx >> 12) % 3;
    const float N = (float)(BB * HH * WW);
    float mean = stats6[c] / N;
    float var = stats6[3 + c] / N - mean * mean;
    float v = y[idx];
    float z = gamma[c] * (v - mean) * rsqrtf(var + 1e-5f) + beta[c];
    y[idx] = siluf(z);
}

// ---------------------------------------------------------------- coordinate-descent step (WMMA)
// Unified LDS matrix sAB (16 rows x 64 cols, padded stride):
//   rows 0-2  : imgp channels
//   rows 3-14 : wedges (candidate j, wedge k -> row 3+3j+k)
//   row  15   : zeros (padding row for unused A row 15 / B cols 12-15)
// A fragment row  = lane&15            -> &sAB[m]
// B fragment col n= lane&15 -> wedge row n+3, clamped to zero-row: min(m+3,15)
// -> every fragment is one branch-free ds_load_b64.
template <int I>
__global__ __launch_bounds__(32) void k_step(const float* __restrict__ xr,
                                             float* __restrict__ params) {
    const int patch = blockIdx.x;
    const int b = patch / (HPP * WPP);
    const int rr = patch % (HPP * WPP);
    const int hp = rr / WPP, wp = rr % WPP;
    const int lane = threadIdx.x;

    __shared__ float sAB[16][ABSTR];
    __shared__ float swsum[4][3];
    __shared__ float sD[16][16];
    __shared__ float sloss[4];

    // zero padding row 15 once
    sAB[15][lane] = 0.f;
    sAB[15][lane + 32] = 0.f;

    // load imgp into rows 0-2; accumulate sum over c,pix of imgp^2
    float ic2p = 0.f;
#pragma unroll
    for (int t = 0; t < 2; ++t) {
        int pix = lane + 32 * t;
        int py = pix >> 3, px = pix & 7;
#pragma unroll
        for (int c = 0; c < 3; ++c) {
            float v = xr[((size_t)(b * 3 + c) * HH + (hp + py)) * WW + (wp + px)];
            sAB[c][pix] = v;
            ic2p += v * v;
        }
    }
    float ic2 = wred(ic2p);
    __syncthreads();

    float p0 = params[((b * 5 + 0) * HPP + hp) * WPP + wp];
    float p1 = params[((b * 5 + 1) * HPP + hp) * WPP + wp];
    float p2 = params[((b * 5 + 2) * HPP + hp) * WPP + wp];
    float p3 = params[((b * 5 + 3) * HPP + hp) * WPP + wp];
    float p4 = params[((b * 5 + 4) * HPP + hp) * WPP + wp];

    float bestLoss = 3.4e38f;
    int bestQ = 0;

    const int m = lane & 15;
    const int khalf = lane >> 4;
    const int rB = (m + 3 < 15) ? (m + 3) : 15;  // v_min, branch-free

    for (int gq = 0; gq < 8; ++gq) {
        // ---- wedges for 4 candidates of this group -> rows 3..14
        for (int j = 0; j < 4; ++j) {
            int q = gq * 4 + j;
            float pr = (I < 3) ? ((float)q * (TWO_PI_F / 31.f)) : (-3.f + (float)q * 0.2f);
            float q0 = p0, q1 = p1, q2 = p2, q3 = p3, q4 = p4;
            if (I == 0) q0 += pr;
            if (I == 1) q1 += pr;
            if (I == 2) q2 += pr;
            if (I == 3) q3 += pr;
            if (I == 4) q4 += pr;
            Geom G;
            make_geom(q0, q1, q2, q3, q4, G);
            float s0 = 0.f, s1 = 0.f, s2 = 0.f;
#pragma unroll
            for (int t = 0; t < 2; ++t) {
                int pix = lane + 32 * t;
                int py = pix >> 3, px = pix & 7;
                float xg = -1.f + (float)px * (2.f / 7.f);
                float yg = -1.f + (float)py * (2.f / 7.f);
                float d13, d42;
                dists_at(G, xg, yg, d13, d42);
                float w0, w1, w2;
                wedges_at(d13, d42, w0, w1, w2);
                if (q >= NV) { w0 = 0.f; w1 = 0.f; w2 = 0.f; }
                sAB[3 + 3 * j + 0][pix] = w0;
                sAB[3 + 3 * j + 1][pix] = w1;
                sAB[3 + 3 * j + 2][pix] = w2;
                s0 += w0; s1 += w1; s2 += w2;
            }
            s0 = wred(s0); s1 = wred(s1); s2 = wred(s2);
            if (lane == 0) { swsum[j][0] = s0; swsum[j][1] = s1; swsum[j][2] = s2; }
        }
        __syncthreads();

        // ---- D = A(16x64) * B(64x16) via 16 chained f32 WMMAs (K=4 each)
        v8f acc = { 0.f, 0.f, 0.f, 0.f, 0.f, 0.f, 0.f, 0.f };
#pragma unroll
        for (int kc = 0; kc < 16; ++kc) {
            const int kb = kc * 4 + 2 * khalf;  // per ISA 16x4 f32 layout
            v2f av = *(const v2f*)&sAB[m][kb];
            v2f bv = *(const v2f*)&sAB[rB][kb];
            acc = __builtin_amdgcn_wmma_f32_16x16x4_f32(false, av, false, bv,
                                                        (short)0, acc, false, false);
        }
#pragma unroll
        for (int v = 0; v < 8; ++v) sD[v + 8 * khalf][m] = acc[v];
        __syncthreads();

        // ---- per-candidate loss from cnum + Gram
        if (lane < 4 && (gq * 4 + lane) < NV) {
            float loss = ic2;
            const int cb = 3 * lane;       // candidate column base
            const int rb = 3 + 3 * lane;   // candidate Gram row base
#pragma unroll
            for (int c = 0; c < 3; ++c) {
                float cn0 = sD[c][cb + 0], cn1 = sD[c][cb + 1], cn2 = sD[c][cb + 2];
                float co0 = cn0 / (swsum[lane][0] + 1e-10f);
                float co1 = cn1 / (swsum[lane][1] + 1e-10f);
                float co2 = cn2 / (swsum[lane][2] + 1e-10f);
                loss -= 2.f * (co0 * cn0 + co1 * cn1 + co2 * cn2);
                float g00 = sD[rb + 0][cb + 0], g01 = sD[rb + 0][cb + 1], g02 = sD[rb + 0][cb + 2];
                float g10 = sD[rb + 1][cb + 0], g11 = sD[rb + 1][cb + 1], g12 = sD[rb + 1][cb + 2];
                float g20 = sD[rb + 2][cb + 0], g21 = sD[rb + 2][cb + 1], g22 = sD[rb + 2][cb + 2];
                loss += co0 * (g00 * co0 + g01 * co1 + g02 * co2)
                      + co1 * (g10 * co0 + g11 * co1 + g12 * co2)
                      + co2 * (g20 * co0 + g21 * co1 + g22 * co2);
            }
            sloss[lane] = loss * (1.f / 64.f);
        }
        __syncthreads();
        for (int j = 0; j < 4; ++j) {
            int q = gq * 4 + j;
            if (q < NV) {
                float L = sloss[j];
                if (L < bestLoss) { bestLoss = L; bestQ = q; }
            }
        }
        __syncthreads();
    }

    if (lane == 0) {
        float pr = (I < 3) ? ((float)bestQ * (TWO_PI_F / 31.f)) : (-3.f + (float)bestQ * 0.2f);
        float base = (I == 0) ? p0 : ((I == 1) ? p1 : ((I == 2) ? p2 : ((I == 3) ? p3 : p4)));
        params[((b * 5 + I) * HPP + hp) * WPP + wp] = base + pr;
    }
}

// ---------------------------------------------------------------- finalize: store geometry + colors per patch
__global__ __launch_bounds__(32) void k_final(const float* __restrict__ xr,
                                              const float* __restrict__ params,
                                              float* __restrict__ fin) {
    const int patch = blockIdx.x;
    const int b = patch / (HPP * WPP);
    const int rr = patch % (HPP * WPP);
    const int hp = rr / WPP, wp = rr % WPP;
    const int lane = threadIdx.x;

    float p0 = params[((b * 5 + 0) * HPP + hp) * WPP + wp];
    float p1 = params[((b * 5 + 1) * HPP + hp) * WPP + wp];
    float p2 = params[((b * 5 + 2) * HPP + hp) * WPP + wp];
    float p3 = params[((b * 5 + 3) * HPP + hp) * WPP + wp];
    float p4 = params[((b * 5 + 4) * HPP + hp) * WPP + wp];
    Geom G;
    make_geom(p0, p1, p2, p3, p4, G);

    float wsum[3] = { 0.f, 0.f, 0.f };
    float cn[3][3] = { { 0.f, 0.f, 0.f }, { 0.f, 0.f, 0.f }, { 0.f, 0.f, 0.f } };
#pragma unroll
    for (int t = 0; t < 2; ++t) {
        int pix = lane + 32 * t;
        int py = pix >> 3, px = pix & 7;
        float xg = -1.f + (float)px * (2.f / 7.f);
        float yg = -1.f + (float)py * (2.f / 7.f);
        float d13, d42;
        dists_at(G, xg, yg, d13, d42);
        float w0, w1, w2;
        wedges_at(d13, d42, w0, w1, w2);
        wsum[0] += w0; wsum[1] += w1; wsum[2] += w2;
#pragma unroll
        for (int c = 0; c < 3; ++c) {
            float v = xr[((size_t)(b * 3 + c) * HH + (hp + py)) * WW + (wp + px)];
            cn[c][0] += v * w0;
            cn[c][1] += v * w1;
            cn[c][2] += v * w2;
        }
    }
#pragma unroll
    for (int k = 0; k < 3; ++k) wsum[k] = wred(wsum[k]);
#pragma unroll
    for (int c = 0; c < 3; ++c)
#pragma unroll
        for (int k = 0; k < 3; ++k) cn[c][k] = wred(cn[c][k]);

    if (lane == 0) {
        float* F = fin + (size_t)patch * 24;
        F[0] = G.sa1; F[1] = G.ca1; F[2] = G.sa2; F[3] = G.ca2;
        F[4] = G.sa3; F[5] = G.ca3; F[6] = G.sa4; F[7] = G.ca4;
        F[8] = G.x0;  F[9] = G.y0;  F[10] = G.s13; F[11] = G.s42;
        F[12] = G.gt13; F[13] = G.gt42;
#pragma unroll
        for (int c = 0; c < 3; ++c)
#pragma unroll
            for (int k = 0; k < 3; ++k)
                F[14 + c * 3 + k] = cn[c][k] / (wsum[k] + 1e-10f);
    }
}

// ---------------------------------------------------------------- gather fold + final conv + BN partials
__global__ void k_gather(const float* __restrict__ fin,
                         const float* __restrict__ fb_w,
                         const float* __restrict__ fi_w,
                         float* __restrict__ simgv, float* __restrict__ sbv,
                         float* __restrict__ part2) {
    int idx = blockIdx.x * 256 + threadIdx.x;  // 0..8191 over (b,h,w)
    int b = idx >> 12;
    int h = (idx >> 6) & 63;
    int w = idx & 63;

    int py0 = max(0, h - (HPP - 1)), py1 = min(PSZ - 1, h);
    int px0 = max(0, w - (WPP - 1)), px1 = min(PSZ - 1, w);

    float acc0 = 0.f, acc1 = 0.f, acc2 = 0.f, accg = 0.f;
    for (int py = py0; py <= py1; ++py) {
        int hp = h - py;
        float yg = -1.f + (float)py * (2.f / 7.f);
        for (int px = px0; px <= px1; ++px) {
            int wp = w - px;
            float xg = -1.f + (float)px * (2.f / 7.f);
            const float* F = fin + (size_t)((b * HPP + hp) * WPP + wp) * 24;
            Geom G;
            G.sa1 = F[0]; G.ca1 = F[1]; G.sa2 = F[2]; G.ca2 = F[3];
            G.sa3 = F[4]; G.ca3 = F[5]; G.sa4 = F[6]; G.ca4 = F[7];
            G.x0 = F[8]; G.y0 = F[9]; G.s13 = F[10]; G.s42 = F[11];
            G.gt13 = F[12]; G.gt42 = F[13];
            float d13, d42;
            dists_at(G, xg, yg, d13, d42);
            float w0, w1, w2;
            wedges_at(d13, d42, w0, w1, w2);
            acc0 += F[14] * w0 + F[15] * w1 + F[16] * w2;
            acc1 += F[17] * w0 + F[18] * w1 + F[19] * w2;
            acc2 += F[20] * w0 + F[21] * w1 + F[22] * w2;
            float mad = (d13 < 0.f) ? -d13
                       : ((d42 < 0.f) ? fminf(d13, -d42) : fminf(d13, d42));
            float r = mad / DELTA_F;
            accg += 1.f / (1.f + r * r);
        }
    }
    float np = (float)((py1 - py0 + 1) * (px1 - px0 + 1));
    float sm0 = acc0 / np, sm1 = acc1 / np, sm2 = acc2 / np;
    float vi = fi_w[0] * sm0 + fi_w[1] * sm1 + fi_w[2] * sm2;
    float vb = (fb_w[0] + fb_w[1] + fb_w[2]) * (accg / np);
    simgv[idx] = vi;
    sbv[idx] = vb;

    __shared__ float red[256];
    float vals[4] = { vi, vi * vi, vb, vb * vb };
#pragma unroll
    for (int s = 0; s < 4; ++s) {
        red[threadIdx.x] = vals[s];
        __syncthreads();
        for (int off = 128; off > 0; off >>= 1) {
            if (threadIdx.x < off) red[threadIdx.x] += red[threadIdx.x + off];
            __syncthreads();
        }
        if (threadIdx.x == 0) part2[blockIdx.x * 4 + s] = red[0];
        __syncthreads();
    }
}

// ---------------------------------------------------------------- final BN + SiLU -> d_out
__global__ void k_out(const float* __restrict__ simgv, const float* __restrict__ sbv,
                      const float* __restrict__ stats4,
                      const float* __restrict__ fb_g, const float* __restrict__ fb_b,
                      const float* __restrict__ fi_g, const float* __restrict__ fi_b,
                      float* __restrict__ out) {
    int idx = blockIdx.x * 256 + threadIdx.x;  // 0..8191
    const float N = (float)(BB * HH * WW);
    float mi = stats4[0] / N, vvi = stats4[1] / N - mi * mi;
    float mb = stats4[2] / N, vvb = stats4[3] / N - mb * mb;
    float zb = fb_g[0] * (sbv[idx] - mb) * rsqrtf(vvb + 1e-5f) + fb_b[0];
    float zi = fi_g[0] * (simgv[idx] - mi) * rsqrtf(vvi + 1e-5f) + fi_b[0];
    out[idx] = siluf(zb);                   // boundaries (B,1,H,W)
    out[BB * HH * WW + idx] = siluf(zi);    // image      (B,1,H,W)
}

// ---------------------------------------------------------------- launcher
extern "C" void kernel_launch(void* const* d_in, const int* in_sizes, int n_in,
                              void* d_out, int out_size, void* d_ws, size_t ws_size,
                              hipStream_t stream) {
    (void)in_sizes; (void)n_in; (void)out_size; (void)ws_size;
    const float* x            = (const float*)d_in[0];
    const float* reduce_w     = (const float*)d_in[1];
    const float* reduce_gamma = (const float*)d_in[2];
    const float* reduce_beta  = (const float*)d_in[3];
    const float* fb_w         = (const float*)d_in[4];
    const float* fb_gamma     = (const float*)d_in[5];
    const float* fb_beta      = (const float*)d_in[6];
    const float* fi_w         = (const float*)d_in[7];
    const float* fi_gamma     = (const float*)d_in[8];
    const float* fi_beta      = (const float*)d_in[9];
    float* out = (float*)d_out;
    float* ws  = (float*)d_ws;

    float* y      = ws;                  // 24576 (conv out, then xr in place)
    float* params = y + 24576;           // 32490
    float* fin    = params + 32490;      // NPATCH*24 = 155952
    float* simgv  = fin + (size_t)NPATCH * 24;   // 8192
    float* sbv    = simgv + 8192;        // 8192
    float* part1  = sbv + 8192;          // 32*8
    float* part2  = part1 + 256;         // 32*4
    float* stats6 = part2 + 128;         // 8
    float* stats4 = stats6 + 8;          // 8

    k_zero<<<dim3((32490 + 255) / 256), dim3(256), 0, stream>>>(params, 32490);
    k_conv<<<dim3(32), dim3(256), 0, stream>>>(x, reduce_w, y, part1);
    k_red<<<dim3(1), dim3(1), 0, stream>>>(part1, stats6, 32, 6, 8);
    k_bnsilu<<<dim3(96), dim3(256), 0, stream>>>(y, stats6, reduce_gamma, reduce_beta);

    k_step<0><<<dim3(NPATCH), dim3(32), 0, stream>>>(y, params);
    k_step<1><<<dim3(NPATCH), dim3(32), 0, stream>>>(y, params);
    k_step<2><<<dim3(NPATCH), dim3(32), 0, stream>>>(y, params);
    k_step<3><<<dim3(NPATCH), dim3(32), 0, stream>>>(y, params);
    k_step<4><<<dim3(NPATCH), dim3(32), 0, stream>>>(y, params);

    k_final<<<dim3(NPATCH), dim3(32), 0, stream>>>(y, params, fin);
    k_gather<<<dim3(32), dim3(256), 0, stream>>>(fin, fb_w, fi_w, simgv, sbv, part2);
    k_red<<<dim3(1), dim3(1), 0, stream>>>(part2, stats4, 32, 4, 4);
    k_out<<<dim3(32), dim3(256), 0, stream>>>(simgv, sbv, stats4,
                                              fb_gamma, fb_beta, fi_gamma, fi_beta, out);
}